// SpectralConv2d_2903397893024
// MI455X (gfx1250) — compile-verified
//
#include <hip/hip_runtime.h>
#include <hip/hip_bf16.h>

typedef __attribute__((ext_vector_type(2))) float v2f;
typedef __attribute__((ext_vector_type(8))) float v8f;

#define BATCH   8
#define CI      32
#define CO      32
#define HH      256
#define WW      256
#define MKY     32   // ky modes
#define NKX     64   // kx modes (32 low + 32 high)
#define MAX_K   512  // largest K over all stages (stage B)

__device__ __forceinline__ float tw_cos(int m) { return cosf(0.02454369260617026f * (float)m); }
__device__ __forceinline__ float tw_sin(int m) { return sinf(0.02454369260617026f * (float)m); }

// ---------------- twiddle builders ----------------

// T1[256][64]: forward row rDFT. col n = c*32+ky. c=0 -> cos, c=1 -> -sin
__global__ void fill_T1(float* __restrict__ T1) {
    int idx = blockIdx.x * blockDim.x + threadIdx.x;
    if (idx >= 256 * 64) return;
    int w = idx >> 6, n = idx & 63, ky = n & 31, c = n >> 5;
    int m = (w * ky) & 255;
    T1[idx] = (c == 0) ? tw_cos(m) : -tw_sin(m);
}

// T2[128][512]: forward column DFT (complex). row j2=2j+cj, col k2=2h+ch.
// Z = sum_h (cos - i sin)(Yr + i Yi):
//   Zr row: [cos, +sin] ; Zi row: [-sin, cos]
__global__ void fill_T2(float* __restrict__ T2) {
    int idx = blockIdx.x * blockDim.x + threadIdx.x;
    if (idx >= 128 * 512) return;
    int j2 = idx >> 9, k2 = idx & 511;
    int j = j2 >> 1, cj = j2 & 1;
    int h = k2 >> 1, ch = k2 & 1;
    int kx = (j < 32) ? j : (192 + j);          // 224..255 for hi corner
    int m = (kx * h) & 255;
    float cs = tw_cos(m), sn = tw_sin(m);
    T2[idx] = (cj == 0) ? ((ch == 0) ? cs : sn)
                        : ((ch == 0) ? -sn : cs);
}

// T3[512][128]: inverse column DFT with 1/H. row h2=2h+ch, col j2=2j+cj.
// tmp = (1/256) sum_j (cos + i sin)(Zr + i Zi):
//   tmpr row: [s*cos, -s*sin] ; tmpi row: [s*sin, s*cos]
__global__ void fill_T3(float* __restrict__ T3) {
    int idx = blockIdx.x * blockDim.x + threadIdx.x;
    if (idx >= 512 * 128) return;
    int h2 = idx >> 7, j2 = idx & 127;
    int h = h2 >> 1, ch = h2 & 1;
    int j = j2 >> 1, cj = j2 & 1;
    int kx = (j < 32) ? j : (192 + j);
    int m = (kx * h) & 255;
    const float s = 1.0f / 256.0f;
    float cs = s * tw_cos(m), sn = s * tw_sin(m);
    T3[idx] = (ch == 0) ? ((cj == 0) ? cs : -sn)
                        : ((cj == 0) ? sn : cs);
}

// T4[64][256]: inverse row rFFT with 1/W and Hermitian doubling.
// out[w] = (1/256)(Re X0 + 2 sum_{k>=1} (Xr cos - Xi sin))
// row k<32: s_k*cos ; row 32+k: -s_k*sin  (s_0=1/256, s_k=2/256)
__global__ void fill_T4(float* __restrict__ T4) {
    int idx = blockIdx.x * blockDim.x + threadIdx.x;
    if (idx >= 64 * 256) return;
    int k = idx >> 8, w = idx & 255;
    int ky = k & 31, c = k >> 5;
    int m = (ky * w) & 255;
    float s = ((ky == 0) ? 1.0f : 2.0f) / 256.0f;
    T4[idx] = (c == 0) ? s * tw_cos(m) : -s * tw_sin(m);
}

// ---------------- generic batched WMMA f32 GEMM ----------------
// C[batch][M][N] = A[batch][M][K] * B[batch][K][N] (+ bias[(row/biasDiv)%biasMod])
// Block = 8 waves (32x8); each wave computes one 16x16 C tile with
// V_WMMA_F32_16X16X4_F32. All 8 waves share the same batch and tileN, so the
// K x 16 B-slice is staged into LDS once per block (8x fewer B global loads;
// conflict-free ds_load in the inner loop). A-operand: per-lane row streaming
// (merges to global_load_b64).
// A lane layout (ISA 7.12.2, 32-bit A 16x4): lanes 0-15 carry M=lane,
// VGPR0=K{0|2}, VGPR1=K{1|3} (low|high half-wave). B 4x16 mirrors the K
// pairing. C: VGPR r -> row r (lanes 0-15) / row r+8 (lanes 16-31).
__global__ void __launch_bounds__(256)
wmma_gemm_f32(const float* __restrict__ A,
              const float* __restrict__ B,
              float* __restrict__ C,
              int M, int N, int K,
              int lda, int ldb, int ldc,
              long aStride, long bStride, long cStride,
              const float* __restrict__ bias,
              int biasDiv, int biasMod)
{
    __shared__ float Bs[MAX_K * 16];   // 32 KB

    const int lane = threadIdx.x & 31;
    const int half = lane >> 4;        // 0: K pair {0,1}; 1: K pair {2,3}
    const int l    = lane & 15;
    const int tileN = blockIdx.x;
    const int tileM = blockIdx.y * blockDim.y + threadIdx.y;
    const long batch = blockIdx.z;

    const float* __restrict__ Ab = A + batch * aStride;
    const float* __restrict__ Bb = B + batch * bStride;
    float*       __restrict__ Cb = C + batch * cStride;

    // cooperative stage of the K x 16 B-slice into LDS (before any exit!)
    {
        const int tid = threadIdx.y * 32 + threadIdx.x;
        const int n0  = tileN * 16;
        for (int idx = tid; idx < K * 16; idx += 256) {
            int k = idx >> 4, c = idx & 15;
            Bs[idx] = Bb[(long)k * ldb + n0 + c];
        }
    }
    __syncthreads();

    if (tileM * 16 >= M) return;       // uniform per wave -> EXEC all-ones

    const float* Arow = Ab + (long)(tileM * 16 + l) * lda;

    v8f acc = {};
    for (int k0 = 0; k0 < K; k0 += 4) {
        const int ka = k0 + 2 * half;
        v2f a; a[0] = Arow[ka];          a[1] = Arow[ka + 1];
        v2f b; b[0] = Bs[ka * 16 + l];   b[1] = Bs[(ka + 1) * 16 + l];
        acc = __builtin_amdgcn_wmma_f32_16x16x4_f32(
            /*neg_a=*/false, a, /*neg_b=*/false, b,
            /*c_mod=*/(short)0, acc, /*reuse_a=*/false, /*reuse_b=*/false);
    }

#pragma unroll
    for (int r = 0; r < 8; ++r) {
        const int row = tileM * 16 + r + 8 * half;
        float v = acc[r];
        if (bias) v += bias[(row / biasDiv) % biasMod];
        Cb[(long)row * ldc + tileN * 16 + l] = v;
    }
}

// ---------------- complex channel mixing (tiny: VALU) ----------------
// S2/S3 layout per (b,ch): [64 kx][{re[32],im[32]}]
__global__ void mix_kernel(const float* __restrict__ S2,
                           const float* __restrict__ w1r, const float* __restrict__ w1i,
                           const float* __restrict__ w2r, const float* __restrict__ w2i,
                           float* __restrict__ S3)
{
    int idx = blockIdx.x * blockDim.x + threadIdx.x;
    const int TOT = BATCH * CO * NKX * MKY;   // 524288
    if (idx >= TOT) return;
    int ky = idx & 31;
    int kx = (idx >> 5) & 63;
    int o  = (idx >> 11) & 31;
    int b  = idx >> 16;

    const float* wr; const float* wi; int x;
    if (kx < 32) { wr = w1r; wi = w1i; x = kx; }
    else         { wr = w2r; wi = w2i; x = kx - 32; }

    float ar = 0.f, ai = 0.f;
    const float* Xb = S2 + (long)b * CI * NKX * 64;
    for (int i = 0; i < CI; ++i) {
        const float* Xp = Xb + ((long)i * NKX + kx) * 64;
        float xr = Xp[ky];
        float xi = Xp[32 + ky];
        long widx = (((long)i * CO + o) * 32 + x) * 32 + ky;
        float re = wr[widx], im = wi[widx];
        ar = fmaf(xr, re, fmaf(-xi, im, ar));
        ai = fmaf(xr, im, fmaf( xi, re, ai));
    }
    float* Yp = S3 + (((long)b * CO + o) * NKX + kx) * 64;
    Yp[ky]      = ar;
    Yp[32 + ky] = ai;
}

// ---------------- launch ----------------
extern "C" void kernel_launch(void* const* d_in, const int* in_sizes, int n_in,
                              void* d_out, int out_size, void* d_ws, size_t ws_size,
                              hipStream_t stream)
{
    const float* x    = (const float*)d_in[0];
    const float* w1r  = (const float*)d_in[1];
    const float* w1i  = (const float*)d_in[2];
    const float* w2r  = (const float*)d_in[3];
    const float* w2i  = (const float*)d_in[4];
    const float* bias = (const float*)d_in[5];
    float* out = (float*)d_out;
    float* ws  = (float*)d_ws;

    // workspace offsets (floats)
    const long o_T1 = 0;                      // 256*64      = 16384
    const long o_T2 = o_T1 + 256 * 64;        // 128*512     = 65536
    const long o_T3 = o_T2 + 128 * 512;       // 512*128     = 65536
    const long o_T4 = o_T3 + 512 * 128;       // 64*256      = 16384
    const long o_S1 = o_T4 + 64 * 256;        // 65536*64    = 4194304  (reused as S4)
    const long o_S2 = o_S1 + (long)65536 * 64;// 8*32*64*64  = 1048576
    const long o_S3 = o_S2 + (long)BATCH * CI * NKX * 64;
    float* T1 = ws + o_T1;  float* T2 = ws + o_T2;
    float* T3 = ws + o_T3;  float* T4 = ws + o_T4;
    float* S1 = ws + o_S1;  float* S2 = ws + o_S2;
    float* S3 = ws + o_S3;  float* S4 = S1;   // S1 dead after stage B

    // twiddles
    fill_T1<<<(256 * 64 + 255) / 256, 256, 0, stream>>>(T1);
    fill_T2<<<(128 * 512 + 255) / 256, 256, 0, stream>>>(T2);
    fill_T3<<<(512 * 128 + 255) / 256, 256, 0, stream>>>(T3);
    fill_T4<<<(64 * 256 + 255) / 256, 256, 0, stream>>>(T4);

    const dim3 blk(32, 8);

    // Stage A: S1[65536][64] = x[65536][256] * T1[256][64]
    {
        dim3 grd(64 / 16, (65536 / 16) / 8, 1);
        wmma_gemm_f32<<<grd, blk, 0, stream>>>(x, T1, S1,
            65536, 64, 256, 256, 64, 64, 0, 0, 0, nullptr, 1, 1);
    }
    // Stage B: per (b,i): S2[128][32] = T2[128][512] * S1blk[512][32]
    {
        dim3 grd(32 / 16, 1, BATCH * CI);
        wmma_gemm_f32<<<grd, blk, 0, stream>>>(T2, S1, S2,
            128, 32, 512, 512, 32, 32,
            0, (long)512 * 32, (long)128 * 32, nullptr, 1, 1);
    }
    // Stage C: complex channel mixing
    {
        const int TOT = BATCH * CO * NKX * MKY;
        mix_kernel<<<(TOT + 255) / 256, 256, 0, stream>>>(S2, w1r, w1i, w2r, w2i, S3);
    }
    // Stage D: per (b,o): S4[512][32] = T3[512][128] * S3blk[128][32]
    {
        dim3 grd(32 / 16, (512 / 16) / 8, BATCH * CO);
        wmma_gemm_f32<<<grd, blk, 0, stream>>>(T3, S3, S4,
            512, 32, 128, 128, 32, 32,
            0, (long)128 * 32, (long)512 * 32, nullptr, 1, 1);
    }
    // Stage E: out[65536][256] = S4[65536][64] * T4[64][256] + bias[o]
    {
        dim3 grd(256 / 16, (65536 / 16) / 8, 1);
        wmma_gemm_f32<<<grd, blk, 0, stream>>>(S4, T4, out,
            65536, 256, 64, 64, 256, 256, 0, 0, 0, bias, HH, CO);
    }
}